// GraphV1Emb_79559974191626
// MI455X (gfx1250) — compile-verified
//
#include <hip/hip_runtime.h>
#include <hip/hip_bf16.h>
#include <math.h>

typedef __attribute__((ext_vector_type(2))) float v2f;
typedef __attribute__((ext_vector_type(8))) float v8f;

#define N_NODES  20000
#define N_EDGES  160000
#define N_GRAPHS 64
#define EMBED_DIM 32
#define IN_GNN   36
#define H1       2048
#define H2       1024
#define HH1      1024
#define HH2      512
#define N_CLASSES 2

// ---------------------------------------------------------------------------
// f32 WMMA GEMM, compile-time N/K: C[M,N] = A[M,K] @ B[K,N] (+bias,+relu opt)
// Block = 8 waves -> 128 rows x 64 cols. B chunk (32 x 64) staged in LDS in a
// k-pair-interleaved layout: element (r,c) at Bs[(r>>1)*160 + 2c + (r&1)], so
// every WMMA B fragment is a single 8B-aligned ds_load_b64 landing directly in
// an adjacent VGPR pair (no mov shuffles), and the half-waves (pair offset 160
// = 32 mod 64 banks) hit disjoint bank halves -> conflict-free.
// Waves past the last M-tile are clamped (not exited) so barriers stay
// uniform; they skip the store.
// ---------------------------------------------------------------------------
template <int N, int K>
__global__ __launch_bounds__(256)
void wmma_gemm_f32(const float* __restrict__ A, const float* __restrict__ B,
                   const float* __restrict__ bias, float* __restrict__ C,
                   int M, int relu) {
  constexpr int KC    = 32;                 // k-chunk
  constexpr int PITCH = 160;                // dwords per k-pair row (128 + 32 pad)
  __shared__ float Bs[(KC / 2) * PITCH];    // 10 KB

  const int lane = threadIdx.x & 31;
  const int wid  = threadIdx.x >> 5;
  const int nStrips = N >> 6;
  const int mTiles  = M >> 4;
  const int strip   = blockIdx.x % nStrips;
  const int mchunk  = blockIdx.x / nStrips;

  int mt = mchunk * 8 + wid;
  const bool active = (mt < mTiles);
  if (!active) mt = mTiles - 1;             // wave-uniform clamp (EXEC stays full)
  const int m0 = mt << 4;
  const int n0 = strip << 6;

  const int half = lane >> 4;               // 0: K {0,1} | 1: K {2,3}
  const int l15  = lane & 15;

  v8f acc0 = {}, acc1 = {}, acc2 = {}, acc3 = {};
  const float* arow = A + (size_t)(m0 + l15) * K + (half << 1);

  for (int kc0 = 0; kc0 < K; kc0 += KC) {
    const int kc = (K - kc0 < KC) ? (K - kc0) : KC;
    __syncthreads();                        // protect previous chunk's reads
    // cooperative fill: thread -> (k-pair p, col c); one ds_store_b64 each
    for (int idx = threadIdx.x; idx < (kc >> 1) * 64; idx += 256) {
      const int p = idx >> 6, c = idx & 63;
      const float* bp = B + (size_t)(kc0 + 2 * p) * N + n0 + c;
      v2f t;
      t.x = bp[0];
      t.y = bp[N];
      *(v2f*)&Bs[p * PITCH + 2 * c] = t;
    }
    __syncthreads();

    const int steps = kc >> 2;
    for (int s = 0; s < steps; ++s) {
      const int k0 = kc0 + (s << 2);
      v2f a;                                // A frag: one global_load_b64
      a.x = arow[k0];
      a.y = arow[k0 + 1];
      __builtin_prefetch(arow + k0 + 8, 0, 3);

      // B frags: one ds_load_b64 each, already in adjacent VGPR pairs
      const float* bl = &Bs[((s << 1) + half) * PITCH + 2 * l15];
      const v2f b0 = *(const v2f*)(bl);
      const v2f b1 = *(const v2f*)(bl + 32);
      const v2f b2 = *(const v2f*)(bl + 64);
      const v2f b3 = *(const v2f*)(bl + 96);

      acc0 = __builtin_amdgcn_wmma_f32_16x16x4_f32(false, a, false, b0, (short)0, acc0, false, false);
      acc1 = __builtin_amdgcn_wmma_f32_16x16x4_f32(false, a, false, b1, (short)0, acc1, false, false);
      acc2 = __builtin_amdgcn_wmma_f32_16x16x4_f32(false, a, false, b2, (short)0, acc2, false, false);
      acc3 = __builtin_amdgcn_wmma_f32_16x16x4_f32(false, a, false, b3, (short)0, acc3, false, false);
    }
  }

  if (!active) return;                      // after last barrier: safe
  v8f* accs[4] = {&acc0, &acc1, &acc2, &acc3};
  #pragma unroll
  for (int t = 0; t < 4; ++t) {
    const int col = n0 + t * 16 + l15;
    const float bv = bias ? bias[col] : 0.0f;
    #pragma unroll
    for (int r = 0; r < 8; ++r) {
      const int row = m0 + r + (half << 3);
      float v = (*accs[t])[r] + bv;
      if (relu) v = fmaxf(v, 0.0f);
      C[(size_t)row * N + col] = v;
    }
  }
}

// ---------------------------------------------------------------------------
// Degree / norm
// ---------------------------------------------------------------------------
__global__ void init_deg_k(float* deg, int n) {
  int i = blockIdx.x * blockDim.x + threadIdx.x;
  if (i < n) deg[i] = 1.0f;   // self-loop contribution
}
__global__ void edge_deg_k(const int* __restrict__ dst, float* deg, int e) {
  int i = blockIdx.x * blockDim.x + threadIdx.x;
  if (i < e) atomicAdd(&deg[dst[i]], 1.0f);
}
__global__ void dinv_k(float* d, int n) {
  int i = blockIdx.x * blockDim.x + threadIdx.x;
  if (i < n) { float v = d[i]; d[i] = (v > 0.0f) ? rsqrtf(v) : 0.0f; }
}

// ---------------------------------------------------------------------------
// x = concat(emb[ids], other_feats)  -> [N, 36]
// ---------------------------------------------------------------------------
__global__ void gather_x_k(const int* __restrict__ ids,
                           const float* __restrict__ other,
                           const float* __restrict__ emb,
                           float* __restrict__ x, int n) {
  int idx = blockIdx.x * blockDim.x + threadIdx.x;
  if (idx >= n * IN_GNN) return;
  int i = idx / IN_GNN, c = idx - i * IN_GNN;
  x[idx] = (c < EMBED_DIM) ? emb[(size_t)ids[i] * EMBED_DIM + c]
                           : other[(size_t)i * 4 + (c - EMBED_DIM)];
}

// ---------------------------------------------------------------------------
// GCN aggregation: h[i] = dinv[i]^2 * xw[i]  (self loop) ...
// ---------------------------------------------------------------------------
__global__ void selfloop_init_k(const float* __restrict__ xw,
                                const float* __restrict__ dinv,
                                float* __restrict__ h, int n, int C) {
  size_t idx = (size_t)blockIdx.x * blockDim.x + threadIdx.x;
  if (idx >= (size_t)n * C) return;
  int i = (int)(idx / C);
  float dv = dinv[i];
  h[idx] = dv * dv * xw[idx];
}
// ... plus edge messages: h[dst] += dinv[src]*dinv[dst] * xw[src]
__global__ void edge_scatter_k(const float* __restrict__ xw,
                               const int* __restrict__ src,
                               const int* __restrict__ dst,
                               const float* __restrict__ dinv,
                               float* __restrict__ h, int e, int C) {
  int ei = blockIdx.x;
  if (ei >= e) return;
  int s = src[ei], d = dst[ei];
  float norm = dinv[s] * dinv[d];
  const float* xr = xw + (size_t)s * C;
  float* hr = h + (size_t)d * C;
  for (int c = threadIdx.x; c < C; c += blockDim.x)
    atomicAdd(&hr[c], norm * xr[c]);
}
__global__ void bias_relu_k(float* __restrict__ h, const float* __restrict__ b,
                            int n, int C) {
  size_t idx = (size_t)blockIdx.x * blockDim.x + threadIdx.x;
  if (idx >= (size_t)n * C) return;
  h[idx] = fmaxf(h[idx] + b[idx % C], 0.0f);
}

// ---------------------------------------------------------------------------
// Mean pool over graphs
// ---------------------------------------------------------------------------
__global__ void zero_f_k(float* p, size_t n) {
  size_t i = (size_t)blockIdx.x * blockDim.x + threadIdx.x;
  if (i < n) p[i] = 0.0f;
}
__global__ void pool_scatter_k(const float* __restrict__ h,
                               const int* __restrict__ batch,
                               float* __restrict__ pooled,
                               float* __restrict__ counts, int n, int C) {
  int i = blockIdx.x;
  if (i >= n) return;
  int g = batch[i];
  if (threadIdx.x == 0) atomicAdd(&counts[g], 1.0f);
  const float* hr = h + (size_t)i * C;
  float* pr = pooled + (size_t)g * C;
  for (int c = threadIdx.x; c < C; c += blockDim.x)
    atomicAdd(&pr[c], hr[c]);
}
__global__ void pool_div_k(float* pooled, const float* counts, int C) {
  int idx = blockIdx.x * blockDim.x + threadIdx.x;
  if (idx >= N_GRAPHS * C) return;
  pooled[idx] /= fmaxf(counts[idx / C], 1.0f);
}

// ---------------------------------------------------------------------------
// Final tiny layer: out[64,2] = hh2[64,512] @ oW[512,2] + ob
// ---------------------------------------------------------------------------
__global__ void final_out_k(const float* __restrict__ h,
                            const float* __restrict__ oW,
                            const float* __restrict__ ob,
                            float* __restrict__ out) {
  int t = blockIdx.x * blockDim.x + threadIdx.x;
  if (t >= N_GRAPHS * N_CLASSES) return;
  int g = t / N_CLASSES, c = t - g * N_CLASSES;
  const float* hr = h + (size_t)g * HH2;
  float acc = ob[c];
  for (int k = 0; k < HH2; ++k) acc = fmaf(hr[k], oW[(size_t)k * N_CLASSES + c], acc);
  out[t] = acc;
}

// ---------------------------------------------------------------------------
// Launch
// ---------------------------------------------------------------------------
static inline int cdiv(long long a, long long b) { return (int)((a + b - 1) / b); }

extern "C" void kernel_launch(void* const* d_in, const int* in_sizes, int n_in,
                              void* d_out, int out_size, void* d_ws, size_t ws_size,
                              hipStream_t stream) {
  const int*   ids   = (const int*)  d_in[0];
  const float* other = (const float*)d_in[1];
  const int*   eidx  = (const int*)  d_in[2];
  const int*   batch = (const int*)  d_in[3];
  const float* emb   = (const float*)d_in[4];
  const float* W1    = (const float*)d_in[5];
  const float* b1    = (const float*)d_in[6];
  const float* W2    = (const float*)d_in[7];
  const float* b2    = (const float*)d_in[8];
  const float* hW1   = (const float*)d_in[9];
  const float* hb1   = (const float*)d_in[10];
  const float* hW2   = (const float*)d_in[11];
  const float* hb2   = (const float*)d_in[12];
  const float* oW    = (const float*)d_in[13];
  const float* ob    = (const float*)d_in[14];
  float* out = (float*)d_out;

  const int* src = eidx;            // edge_index[0]
  const int* dst = eidx + N_EDGES;  // edge_index[1]

  // ---- workspace layout (bytes) ----
  const size_t SZ_BUF = (size_t)N_NODES * H1 * sizeof(float);        // 163.84 MB
  char* ws = (char*)d_ws;
  float* bufA   = (float*)(ws);                                      // xw1 / xw2
  float* bufB   = (float*)(ws + SZ_BUF);                             // h1 / h2
  float* xfeat  = (float*)(ws + 2 * SZ_BUF);                         // [N,36]
  size_t off    = 2 * SZ_BUF + (size_t)N_NODES * IN_GNN * sizeof(float);
  float* dinv   = (float*)(ws + off);  off += ((size_t)N_NODES * 4 + 127) & ~(size_t)127;
  float* pooled = (float*)(ws + off);  off += (size_t)N_GRAPHS * H2 * 4;
  float* counts = (float*)(ws + off);  off += 256;
  float* hh1    = (float*)(ws + off);  off += (size_t)N_GRAPHS * HH1 * 4;
  float* hh2    = (float*)(ws + off);

  // ---- 1. degree / dinv ----
  init_deg_k<<<cdiv(N_NODES, 256), 256, 0, stream>>>(dinv, N_NODES);
  edge_deg_k<<<cdiv(N_EDGES, 256), 256, 0, stream>>>(dst, dinv, N_EDGES);
  dinv_k<<<cdiv(N_NODES, 256), 256, 0, stream>>>(dinv, N_NODES);

  // ---- 2. feature gather ----
  gather_x_k<<<cdiv((long long)N_NODES * IN_GNN, 256), 256, 0, stream>>>(
      ids, other, emb, xfeat, N_NODES);

  // ---- 3. GCN layer 1:  xw1 = x @ W1 ----
  {
    int blocks = cdiv(N_NODES / 16, 8) * (H1 / 64);
    wmma_gemm_f32<H1, IN_GNN><<<blocks, 256, 0, stream>>>(
        xfeat, W1, nullptr, bufA, N_NODES, 0);
  }
  selfloop_init_k<<<cdiv((long long)N_NODES * H1, 256), 256, 0, stream>>>(
      bufA, dinv, bufB, N_NODES, H1);
  edge_scatter_k<<<N_EDGES, 256, 0, stream>>>(bufA, src, dst, dinv, bufB, N_EDGES, H1);
  bias_relu_k<<<cdiv((long long)N_NODES * H1, 256), 256, 0, stream>>>(
      bufB, b1, N_NODES, H1);

  // ---- 4. GCN layer 2:  xw2 = h1 @ W2 ----
  {
    int blocks = cdiv(N_NODES / 16, 8) * (H2 / 64);
    wmma_gemm_f32<H2, H1><<<blocks, 256, 0, stream>>>(
        bufB, W2, nullptr, bufA, N_NODES, 0);
  }
  selfloop_init_k<<<cdiv((long long)N_NODES * H2, 256), 256, 0, stream>>>(
      bufA, dinv, bufB, N_NODES, H2);
  edge_scatter_k<<<N_EDGES, 256, 0, stream>>>(bufA, src, dst, dinv, bufB, N_EDGES, H2);
  bias_relu_k<<<cdiv((long long)N_NODES * H2, 256), 256, 0, stream>>>(
      bufB, b2, N_NODES, H2);

  // ---- 5. mean pool ----
  zero_f_k<<<cdiv((long long)N_GRAPHS * H2, 256), 256, 0, stream>>>(
      pooled, (size_t)N_GRAPHS * H2);
  zero_f_k<<<1, 256, 0, stream>>>(counts, (size_t)N_GRAPHS);
  pool_scatter_k<<<N_NODES, 256, 0, stream>>>(bufB, batch, pooled, counts, N_NODES, H2);
  pool_div_k<<<cdiv((long long)N_GRAPHS * H2, 256), 256, 0, stream>>>(pooled, counts, H2);

  // ---- 6. MLP head ----
  {
    int blocks = cdiv(N_GRAPHS / 16, 8) * (HH1 / 64);   // 1 * 16
    wmma_gemm_f32<HH1, H2><<<blocks, 256, 0, stream>>>(
        pooled, hW1, hb1, hh1, N_GRAPHS, 1);
  }
  {
    int blocks = cdiv(N_GRAPHS / 16, 8) * (HH2 / 64);   // 1 * 8
    wmma_gemm_f32<HH2, HH1><<<blocks, 256, 0, stream>>>(
        hh1, hW2, hb2, hh2, N_GRAPHS, 1);
  }
  final_out_k<<<1, 128, 0, stream>>>(hh2, oW, ob, out);
}